// EdgeMLPEncoder_91156385890469
// MI455X (gfx1250) — compile-verified
//
#include <hip/hip_runtime.h>
#include <math.h>

typedef __attribute__((ext_vector_type(2))) float v2f;
typedef __attribute__((ext_vector_type(8))) float v8f;

#define H       128     // hidden width (fixed by model)
#define CHUNK   64      // rows of L processed per inner pass
#define LDST    132     // padded LDS row stride (floats): 132 % 64 == 4 -> conflict-free A frags
#define NEG_BIG (-1.0e30f)

__device__ __forceinline__ float gelu_exact(float v) {
    // matches jax.nn.gelu(approximate=False): 0.5*x*(1+erf(x/sqrt(2)))
    return 0.5f * v * (1.0f + erff(v * 0.70710678118654752f));
}

// ---------------------------------------------------------------------------
// Stage 1: per block = 1 batch-slice of rows.
//   dx -> gelu(dx@W1+b1) -> gelu(h1@W2+b2) via V_WMMA_F32_16X16X4_F32,
//   reduce sum & max over rows, write per-block partials [H] to workspace.
// 256 threads = 8 waves; wave w owns output N-tile w (cols 16w..16w+15) and
// keeps all 32 W2 B-fragments resident in VGPRs.
// ---------------------------------------------------------------------------
__global__ __launch_bounds__(256)
void edge_mlp_stage1(const float* __restrict__ x,
                     const float* __restrict__ W1,
                     const float* __restrict__ b1,
                     const float* __restrict__ W2,
                     const float* __restrict__ b2,
                     float* __restrict__ wsSum,
                     float* __restrict__ wsMax,
                     int L, int rowsPerBlock, int blocksPerB)
{
    __shared__ float h1s[CHUNK * LDST];   // gelu(dx@W1+b1), padded rows
    __shared__ float dxs[CHUNK * 2];
    __shared__ float w1s0[H], w1s1[H], b1s[H];

    const int tid  = threadIdx.x;
    const int lane = tid & 31;
    const int wv   = tid >> 5;            // 0..7 -> N-tile id
    const int half = lane >> 4;           // 0: K pair {0,1}, 1: K pair {2,3}
    const int m16  = lane & 15;

    const int b    = blockIdx.x / blocksPerB;
    const int blk  = blockIdx.x % blocksPerB;
    const long rowStart = (long)blk * rowsPerBlock;
    const int  chunks   = rowsPerBlock / CHUNK;
    const float* __restrict__ xb = x + (size_t)b * (size_t)L * 2;

    if (tid < H) {
        w1s0[tid] = W1[tid];              // W1[0][c]
        w1s1[tid] = W1[H + tid];          // W1[1][c]
        b1s[tid]  = b1[tid];
    }

    // Persistent B fragments for this wave's N-tile: W2 is [H][H] row-major.
    // B frag (4x16) at K-step kk: lane<16 -> v0=W2[4kk+0][col], v1=W2[4kk+1][col]
    //                             lane>=16-> v0=W2[4kk+2][col], v1=W2[4kk+3][col]
    const int colB = wv * 16 + m16;
    v2f bfr[H / 4];
#pragma unroll
    for (int kk = 0; kk < H / 4; ++kk) {
        const int k0 = 4 * kk + 2 * half;
        v2f t;
        t.x = W2[(size_t)k0 * H + colB];
        t.y = W2[(size_t)(k0 + 1) * H + colB];
        bfr[kk] = t;
    }
    const float bias2 = b2[colB];

    float sumAcc = 0.0f;
    float maxAcc = NEG_BIG;

    __syncthreads();

    for (int ch = 0; ch < chunks; ++ch) {
        const long r0 = rowStart + (long)ch * CHUNK;

        // ---- dx for this chunk (edge diff with wrap along L) ----
        if (tid < CHUNK) {
            const long r  = r0 + tid;
            const long rn = (r + 1 == (long)L) ? 0 : (r + 1);
            dxs[tid * 2 + 0] = xb[rn * 2 + 0] - xb[r * 2 + 0];
            dxs[tid * 2 + 1] = xb[rn * 2 + 1] - xb[r * 2 + 1];
        }
        __syncthreads();

        // ---- layer 1: h1[r][c] = gelu(dx@W1 + b1), each thread one (col, row-half) ----
        {
            const int c  = tid & (H - 1);
            const int rh = tid >> 7;              // 0 or 1 -> rows [32rh, 32rh+32)
            const float w0 = w1s0[c], w1 = w1s1[c], bb = b1s[c];
            for (int r = rh * 32; r < rh * 32 + 32; ++r) {
                const float v = fmaf(dxs[r * 2 + 0], w0,
                                fmaf(dxs[r * 2 + 1], w1, bb));
                h1s[r * LDST + c] = gelu_exact(v);
            }
        }
        __syncthreads();

        // ---- layer 2: WMMA f32 16x16x4, K = 128 (32 steps), 4 M-tiles ----
#pragma unroll
        for (int mt = 0; mt < CHUNK / 16; ++mt) {
            v8f acc = {};
            const float* aBase = &h1s[(mt * 16 + m16) * LDST + 2 * half];
#pragma unroll
            for (int kk = 0; kk < H / 4; ++kk) {
                v2f a = *(const v2f*)(aBase + 4 * kk);
                acc = __builtin_amdgcn_wmma_f32_16x16x4_f32(
                        false, a, false, bfr[kk], (short)0, acc, false, false);
            }
            // acc: lane holds column colB; VGPR r -> row (mt*16 + r + 8*half).
#pragma unroll
            for (int r = 0; r < 8; ++r) {
                const float h2 = gelu_exact(acc[r] + bias2);
                sumAcc += h2;
                maxAcc  = fmaxf(maxAcc, h2);
            }
        }
        __syncthreads();   // protect dxs/h1s before next chunk
    }

    // Combine lane L with L+16 (same column, rows 0-7 vs 8-15).
    sumAcc += __shfl_xor(sumAcc, 16, 32);
    maxAcc  = fmaxf(maxAcc, __shfl_xor(maxAcc, 16, 32));

    if (lane < 16) {
        const size_t base = (size_t)blockIdx.x * H;
        wsSum[base + colB] = sumAcc;
        wsMax[base + colB] = maxAcc;
    }
}

// ---------------------------------------------------------------------------
// Stage 2: per batch, reduce partials, z = [mean||max] @ Wh + bh, LayerNorm.
// ---------------------------------------------------------------------------
__global__ __launch_bounds__(128)
void edge_mlp_stage2(const float* __restrict__ wsSum,
                     const float* __restrict__ wsMax,
                     const float* __restrict__ Wh,
                     const float* __restrict__ bh,
                     const float* __restrict__ gamma,
                     const float* __restrict__ beta,
                     float* __restrict__ out,
                     int L, int nPart)
{
    __shared__ float meanS[H], maxS[H], red[H];
    const int b = blockIdx.x;
    const int o = threadIdx.x;

    // fixed-order reduction over partials -> deterministic
    float s = 0.0f, m = NEG_BIG;
    const size_t base = (size_t)b * nPart * H;
    for (int p = 0; p < nPart; ++p) {
        s += wsSum[base + (size_t)p * H + o];
        m  = fmaxf(m, wsMax[base + (size_t)p * H + o]);
    }
    meanS[o] = s * (1.0f / (float)L);
    maxS[o]  = m;
    __syncthreads();

    float z = bh[o];
    for (int k = 0; k < H; ++k) z = fmaf(meanS[k], Wh[(size_t)k * H + o], z);
    for (int k = 0; k < H; ++k) z = fmaf(maxS[k],  Wh[(size_t)(H + k) * H + o], z);

    // LayerNorm over the 128 outputs (population variance, eps = 1e-5)
    red[o] = z;
    __syncthreads();
    for (int st = H / 2; st > 0; st >>= 1) {
        if (o < st) red[o] += red[o + st];
        __syncthreads();
    }
    const float mu = red[0] * (1.0f / (float)H);
    __syncthreads();
    const float d = z - mu;
    red[o] = d * d;
    __syncthreads();
    for (int st = H / 2; st > 0; st >>= 1) {
        if (o < st) red[o] += red[o + st];
        __syncthreads();
    }
    const float var = red[0] * (1.0f / (float)H);
    out[(size_t)b * H + o] = d * rsqrtf(var + 1e-5f) * gamma[o] + beta[o];
}

// ---------------------------------------------------------------------------
extern "C" void kernel_launch(void* const* d_in, const int* in_sizes, int n_in,
                              void* d_out, int out_size, void* d_ws, size_t ws_size,
                              hipStream_t stream)
{
    const float* x     = (const float*)d_in[0];
    const float* W1    = (const float*)d_in[1];
    const float* b1    = (const float*)d_in[2];
    const float* W2    = (const float*)d_in[3];
    const float* b2    = (const float*)d_in[4];
    const float* Wh    = (const float*)d_in[5];
    const float* bh    = (const float*)d_in[6];
    const float* gamma = (const float*)d_in[7];
    const float* beta  = (const float*)d_in[8];
    float* out = (float*)d_out;

    const int B = out_size / H;                       // 64
    const long L = (long)in_sizes[0] / (2L * B);      // 16384

    // choose per-batch split so rowsPerBlock is a multiple of CHUNK
    int blocksPerB = 32;
    while (blocksPerB > 1 && (L % ((long)blocksPerB * CHUNK)) != 0) blocksPerB >>= 1;
    const int rowsPerBlock = (int)(L / blocksPerB);

    float* wsSum = (float*)d_ws;
    float* wsMax = wsSum + (size_t)B * blocksPerB * H;

    edge_mlp_stage1<<<dim3(B * blocksPerB), dim3(256), 0, stream>>>(
        x, W1, b1, W2, b2, wsSum, wsMax, (int)L, rowsPerBlock, blocksPerB);

    edge_mlp_stage2<<<dim3(B), dim3(128), 0, stream>>>(
        wsSum, wsMax, Wh, bh, gamma, beta, out, (int)L, blocksPerB);
}